// OrbitGNN_40097814675825
// MI455X (gfx1250) — compile-verified
//
#include <hip/hip_runtime.h>
#include <cmath>

typedef __attribute__((ext_vector_type(2))) float v2f;
typedef __attribute__((ext_vector_type(8))) float v8f;

// ---------------- helpers ----------------
__device__ __forceinline__ float lrelu(float x) { return x >= 0.f ? x : 0.2f * x; }

// monotone float atomic-max via signed/unsigned int atomics (works for +/- values)
__device__ __forceinline__ void atomicMaxFloat(float* addr, float val) {
    if (val >= 0.f) atomicMax((int*)addr, __float_as_int(val));
    else            atomicMin((unsigned int*)addr, __float_as_uint(val));
}

// ---------------- dense: scalar linear (layer-1, K=7) ----------------
// out[n, o] = sum_k x[n*Kin+k] * W[o*Kin+k]
__global__ void k_lin_small(const float* __restrict__ x, const float* __restrict__ W,
                            float* __restrict__ out, int N, int Kin, int Nout) {
    int idx = blockIdx.x * blockDim.x + threadIdx.x;
    if (idx >= N * Nout) return;
    int n = idx / Nout, o = idx % Nout;
    const float* xr = x + (size_t)n * Kin;
    const float* wr = W + (size_t)o * Kin;
    float s = 0.f;
    for (int k = 0; k < Kin; ++k) s += xr[k] * wr[k];
    out[idx] = s;
}

// ---------------- dense: WMMA fp32 GEMM  C[N,NOUT] = A[N,K] @ W^T, W:[NOUT,K] ----------------
// One wave per 16-row tile, NT=NOUT/16 accumulators. B fragments for all column
// tiles are staged into distinct registers per k-step so all loads issue before
// the WMMA burst (avoids load->wait->wmma serialization seen in round 0).
template <int K, int NOUT, bool BIAS_RELU>
__global__ void wmma_gemm(const float* __restrict__ A, const float* __restrict__ W,
                          const float* __restrict__ bias, float* __restrict__ out, int nrows) {
    constexpr int NT = NOUT / 16;
    int wave = (blockIdx.x * blockDim.x + threadIdx.x) >> 5;
    int lane = threadIdx.x & 31;
    int row0 = wave * 16;
    if (row0 >= nrows) return;
    int l15 = lane & 15;
    int kh  = lane >> 4;                 // 0 or 1 -> selects K pair {0,1} vs {2,3}
    v8f acc[NT];
    #pragma unroll
    for (int t = 0; t < NT; ++t) acc[t] = {};
    int arow_i = row0 + l15; if (arow_i >= nrows) arow_i = nrows - 1; // safety clamp
    const float* arow = A + (size_t)arow_i * K + kh * 2;
    const float* wrow = W + (size_t)l15 * K + kh * 2;
    #pragma unroll 2
    for (int kt = 0; kt < K; kt += 4) {
        v2f a = *(const v2f*)(arow + kt);
        v2f b[NT];
        #pragma unroll
        for (int t = 0; t < NT; ++t)
            b[t] = *(const v2f*)(wrow + (size_t)t * 16 * K + kt);
        #pragma unroll
        for (int t = 0; t < NT; ++t)
            acc[t] = __builtin_amdgcn_wmma_f32_16x16x4_f32(
                false, a, false, b[t], (short)0, acc[t], false, false);
    }
    #pragma unroll
    for (int t = 0; t < NT; ++t) {
        int col = t * 16 + l15;
        #pragma unroll
        for (int i = 0; i < 8; ++i) {
            int row = row0 + kh * 8 + i;
            if (row >= nrows) continue;
            float v = acc[t][i];
            if (BIAS_RELU) { v += bias[col]; v = v > 0.f ? v : 0.f; }
            out[(size_t)row * NOUT + col] = v;
        }
    }
}

// ---------------- attention coefficients + per-node init ----------------
// alpha_s/alpha_d per (node, head); also init segment-max to -inf and denom to 0
__global__ void k_alpha(const float* __restrict__ h, const float* __restrict__ a_src,
                        const float* __restrict__ a_dst, float* __restrict__ as_,
                        float* __restrict__ ad_, float* __restrict__ mx, float* __restrict__ dn,
                        int N) {
    int idx = blockIdx.x * blockDim.x + threadIdx.x;
    if (idx >= N * 2) return;
    int n = idx >> 1, hh = idx & 1;
    const float* hr = h + (size_t)n * 128 + hh * 64;
    const float* sr = a_src + hh * 64;
    const float* dr = a_dst + hh * 64;
    float s = 0.f, d = 0.f;
    #pragma unroll 8
    for (int k = 0; k < 64; ++k) { float v = hr[k]; s += v * sr[k]; d += v * dr[k]; }
    as_[idx] = s; ad_[idx] = d;
    mx[idx] = -__builtin_huge_valf();
    dn[idx] = 0.f;
}

__device__ __forceinline__ void edge_pair(const int* __restrict__ ei, int E, int i,
                                          int& src, int& dst) {
    if (i < E) { src = ei[i]; dst = ei[E + i]; }
    else       { src = dst = i - E; }     // self-loops appended
}

// ---------------- edge pass 1: segment max ----------------
__global__ void k_edge_max(const int* __restrict__ ei, int E, int N,
                           const float* __restrict__ as_, const float* __restrict__ ad_,
                           float* __restrict__ mx) {
    int i = blockIdx.x * blockDim.x + threadIdx.x;
    if (i >= E + N) return;
    int src, dst; edge_pair(ei, E, i, src, dst);
    #pragma unroll
    for (int hh = 0; hh < 2; ++hh) {
        float e = lrelu(as_[src * 2 + hh] + ad_[dst * 2 + hh]);
        atomicMaxFloat(&mx[dst * 2 + hh], e);
    }
}

// ---------------- edge pass 2: segment sum of exp ----------------
__global__ void k_edge_sum(const int* __restrict__ ei, int E, int N,
                           const float* __restrict__ as_, const float* __restrict__ ad_,
                           const float* __restrict__ mx, float* __restrict__ dn) {
    int i = blockIdx.x * blockDim.x + threadIdx.x;
    if (i >= E + N) return;
    int src, dst; edge_pair(ei, E, i, src, dst);
    #pragma unroll
    for (int hh = 0; hh < 2; ++hh) {
        float e = lrelu(as_[src * 2 + hh] + ad_[dst * 2 + hh]);
        atomicAdd(&dn[dst * 2 + hh], expf(e - mx[dst * 2 + hh]));
    }
}

// ---------------- edge pass 3: weighted scatter (wave per edge, float4 per lane) ----------------
__global__ void k_edge_aggr(const int* __restrict__ ei, int E, int N,
                            const float* __restrict__ as_, const float* __restrict__ ad_,
                            const float* __restrict__ mx, const float* __restrict__ dn,
                            const float* __restrict__ h, float* __restrict__ out) {
    int wave = (blockIdx.x * blockDim.x + threadIdx.x) >> 5;
    int lane = threadIdx.x & 31;
    if (wave >= E + N) return;
    int src, dst; edge_pair(ei, E, wave, src, dst);
    int c  = lane * 4;       // 0..124 covers both heads' 64 dims
    int hh = c >> 6;         // head index
    float e = lrelu(as_[src * 2 + hh] + ad_[dst * 2 + hh]);
    float w = expf(e - mx[dst * 2 + hh]) / (dn[dst * 2 + hh] + 1e-16f);
    float4 hv = *(const float4*)(h + (size_t)src * 128 + c);
    float* o = out + (size_t)dst * 128 + c;
    atomicAdd(o + 0, hv.x * w);
    atomicAdd(o + 1, hv.y * w);
    atomicAdd(o + 2, hv.z * w);
    atomicAdd(o + 3, hv.w * w);
}

// ---------------- elementwise ----------------
__global__ void k_fill_bias(float* __restrict__ out, const float* __restrict__ b, int total) {
    int i = blockIdx.x * blockDim.x + threadIdx.x;
    if (i < total) out[i] = b[i & 127];
}
__global__ void k_fill_zero(float* __restrict__ out, int total) {
    int i = blockIdx.x * blockDim.x + threadIdx.x;
    if (i < total) out[i] = 0.f;
}
__global__ void k_relu(float* __restrict__ x, int total) {
    int i = blockIdx.x * blockDim.x + threadIdx.x;
    if (i < total) { float v = x[i]; x[i] = v > 0.f ? v : 0.f; }
}
// mean over 2 heads + bias: dst[n*64+d] = 0.5*(src[n*128+d] + src[n*128+64+d]) + b[d]
__global__ void k_mean_bias(const float* __restrict__ src, const float* __restrict__ b,
                            float* __restrict__ dst, int N) {
    int i = blockIdx.x * blockDim.x + threadIdx.x;
    if (i >= N * 64) return;
    int n = i >> 6, d = i & 63;
    dst[i] = 0.5f * (src[(size_t)n * 128 + d] + src[(size_t)n * 128 + 64 + d]) + b[d];
}
// final classifier: out[n,c] = dot64(hc[n,:], Wc2[c,:]) + bc2[c]
__global__ void k_final(const float* __restrict__ hc, const float* __restrict__ Wc2,
                        const float* __restrict__ bc2, float* __restrict__ out, int N) {
    int n = blockIdx.x * blockDim.x + threadIdx.x;
    if (n >= N) return;
    const float* hr = hc + (size_t)n * 64;
    float acc[4] = {bc2[0], bc2[1], bc2[2], bc2[3]};
    for (int k = 0; k < 64; ++k) {
        float v = hr[k];
        #pragma unroll
        for (int cidx = 0; cidx < 4; ++cidx) acc[cidx] += v * Wc2[cidx * 64 + k];
    }
    #pragma unroll
    for (int cidx = 0; cidx < 4; ++cidx) out[(size_t)n * 4 + cidx] = acc[cidx];
}

// ---------------- launcher ----------------
static inline int cdiv(int a, int b) { return (a + b - 1) / b; }

extern "C" void kernel_launch(void* const* d_in, const int* in_sizes, int n_in,
                              void* d_out, int out_size, void* d_ws, size_t ws_size,
                              hipStream_t stream) {
    const float* x      = (const float*)d_in[0];
    const int*   ei     = (const int*)  d_in[1];   // [2, E]
    const float* W1     = (const float*)d_in[2];   // [128, 7]
    const float* a_src1 = (const float*)d_in[3];   // [2, 64]
    const float* a_dst1 = (const float*)d_in[4];
    const float* b1     = (const float*)d_in[5];   // [128]
    const float* W2     = (const float*)d_in[6];   // [128, 128]
    const float* a_src2 = (const float*)d_in[7];
    const float* a_dst2 = (const float*)d_in[8];
    const float* b2     = (const float*)d_in[9];   // [64]
    const float* Wc1    = (const float*)d_in[10];  // [64, 64]
    const float* bc1    = (const float*)d_in[11];
    const float* Wc2    = (const float*)d_in[12];  // [4, 64]
    const float* bc2    = (const float*)d_in[13];
    float* out = (float*)d_out;

    const int N = in_sizes[0] / 7;
    const int E = in_sizes[1] / 2;
    const int ET = E + N;          // edges incl. self-loops

    float* buf1 = (float*)d_ws;            // N*128 (h1, then h2, then h2_mean, scratch)
    float* buf2 = buf1 + (size_t)N * 128;  // N*128 (out1, then out2, then hc)
    float* as_  = buf2 + (size_t)N * 128;  // N*2
    float* ad_  = as_  + (size_t)N * 2;
    float* mx_  = ad_  + (size_t)N * 2;
    float* dn_  = mx_  + (size_t)N * 2;

    const int B = 256;
    const int gNE        = cdiv(ET, B);         // thread-per-edge kernels
    const int waveBlocks = cdiv(ET, B / 32);    // wave-per-edge kernel
    const int g128       = cdiv(N * 128, B);
    const int gemmBlocks = cdiv(cdiv(N, 16) * 32, B);

    // ---------- GAT layer 1 ----------
    k_lin_small<<<g128, B, 0, stream>>>(x, W1, buf1, N, 7, 128);
    k_alpha<<<cdiv(N * 2, B), B, 0, stream>>>(buf1, a_src1, a_dst1, as_, ad_, mx_, dn_, N);
    k_edge_max<<<gNE, B, 0, stream>>>(ei, E, N, as_, ad_, mx_);
    k_edge_sum<<<gNE, B, 0, stream>>>(ei, E, N, as_, ad_, mx_, dn_);
    k_fill_bias<<<g128, B, 0, stream>>>(buf2, b1, N * 128);
    k_edge_aggr<<<waveBlocks, B, 0, stream>>>(ei, E, N, as_, ad_, mx_, dn_, buf1, buf2);
    k_relu<<<g128, B, 0, stream>>>(buf2, N * 128);

    // ---------- GAT layer 2 ----------
    wmma_gemm<128, 128, false><<<gemmBlocks, B, 0, stream>>>(buf2, W2, nullptr, buf1, N);
    k_alpha<<<cdiv(N * 2, B), B, 0, stream>>>(buf1, a_src2, a_dst2, as_, ad_, mx_, dn_, N);
    k_edge_max<<<gNE, B, 0, stream>>>(ei, E, N, as_, ad_, mx_);
    k_edge_sum<<<gNE, B, 0, stream>>>(ei, E, N, as_, ad_, mx_, dn_);
    k_fill_zero<<<g128, B, 0, stream>>>(buf2, N * 128);
    k_edge_aggr<<<waveBlocks, B, 0, stream>>>(ei, E, N, as_, ad_, mx_, dn_, buf1, buf2);
    k_mean_bias<<<cdiv(N * 64, B), B, 0, stream>>>(buf2, b2, buf1, N);

    // ---------- classifier MLP ----------
    wmma_gemm<64, 64, true><<<gemmBlocks, B, 0, stream>>>(buf1, Wc1, bc1, buf2, N);
    k_final<<<cdiv(N, B), B, 0, stream>>>(buf2, Wc2, bc2, out, N);
    (void)n_in; (void)out_size; (void)ws_size;
}